// GNN_node_73710228734481
// MI455X (gfx1250) — compile-verified
//
#include <hip/hip_runtime.h>
#include <hip/hip_bf16.h>

// GIN (5 layers, D=128, F=16) for MI455X / gfx1250, wave32 + WMMA f32 16x16x4.
// Memory-bound on edge gather/scatter (~4.4GB vs 23.3TB/s); agg fits in 192MB L2
// so scatter atomics resolve at L2. All GEMMs use V_WMMA_F32_16X16X4_F32.
// Weight staging uses GLOBAL_LOAD_ASYNC_TO_LDS_B128 (ASYNCcnt-tracked DMA to LDS).

#define D_ 128
#define F_ 16
#define L_ 5

#define ZP  132   // LDS pitch for A (activation) tiles: 132%64=4 -> 16 row-lanes hit distinct banks
#define WP1 264   // LDS pitch for W1 (256 cols): 2*264%64=16 -> K and K+2 rows in disjoint bank sets
#define WP2 136   // LDS pitch for W2 (128 cols): 2*136%64=16
#define WPE 136   // LDS pitch for 16x128 encoder weights

typedef __attribute__((ext_vector_type(2))) float v2f;
typedef __attribute__((ext_vector_type(8))) float v8f;

__device__ __forceinline__ v8f wmma_f32(v2f a, v2f b, v8f c) {
  // 8-arg form: (neg_a, A, neg_b, B, c_mod, C, reuse_a, reuse_b)
  return __builtin_amdgcn_wmma_f32_16x16x4_f32(false, a, false, b, (short)0, c,
                                               false, false);
}

// Async global->LDS 16B copy (per-lane), tracked by ASYNCcnt.
__device__ __forceinline__ void async_copy16(void* ldsPtr, const void* gptr) {
  unsigned lws = (unsigned)(size_t)ldsPtr;   // low 32 bits = LDS byte address
  asm volatile("global_load_async_to_lds_b128 %0, %1, off"
               :: "v"(lws), "v"(gptr) : "memory");
}
__device__ __forceinline__ void async_wait0() {
  asm volatile("s_wait_asynccnt 0x0" ::: "memory");
}

// ---------------------------------------------------------------------------
// h = x @ W_node + b_node      [N,16] @ [16,128]
// 8 waves/block, each wave does a 16-node x 128-col tile (8 WMMA col-tiles, 4 K-steps).
// ---------------------------------------------------------------------------
__global__ void __launch_bounds__(256) node_enc_kernel(
    const float* __restrict__ x, const float* __restrict__ Wn,
    const float* __restrict__ bnode, float* __restrict__ h, int N) {
  __shared__ float ldsW[F_ * WPE];
  __shared__ float ldsB[D_];
  for (int i = threadIdx.x; i < F_ * D_; i += 256) {
    int r = i >> 7, c = i & 127;
    ldsW[r * WPE + c] = Wn[i];
  }
  if (threadIdx.x < D_) ldsB[threadIdx.x] = bnode[threadIdx.x];
  __syncthreads();

  int wave = threadIdx.x >> 5, lane = threadIdx.x & 31;
  int lm = lane & 15, dlt = (lane >> 4) << 1;
  int n0 = (blockIdx.x * 8 + wave) * 16;
  if (n0 >= N) return;
  bool full = (n0 + 16 <= N);              // wave-uniform
  int arow = n0 + lm; if (arow >= N) arow = N - 1;

  v2f A[4];
#pragma unroll
  for (int kt = 0; kt < 4; ++kt) {
    int k = kt * 4 + dlt;
    A[kt].x = x[arow * F_ + k];
    A[kt].y = x[arow * F_ + k + 1];
  }
  int mbase = (lane >> 4) << 3;
#pragma unroll
  for (int ct = 0; ct < 8; ++ct) {
    int c = ct * 16 + lm;
    v8f acc = {};
#pragma unroll
    for (int kt = 0; kt < 4; ++kt) {
      int k = kt * 4 + dlt;
      v2f B = { ldsW[k * WPE + c], ldsW[(k + 1) * WPE + c] };
      acc = wmma_f32(A[kt], B, acc);
    }
    float bias = ldsB[c];
    if (full) {
#pragma unroll
      for (int v = 0; v < 8; ++v)
        h[(n0 + mbase + v) * D_ + c] = acc[v] + bias;
    } else {
#pragma unroll
      for (int v = 0; v < 8; ++v) {
        int row = n0 + mbase + v;
        if (row < N) h[row * D_ + c] = acc[v] + bias;
      }
    }
  }
}

// ---------------------------------------------------------------------------
// Fused edge phase: e = edge_attr @ We + be (WMMA, K=16); msg = relu(h[src]+e);
// atomicAdd into agg[dst]. One 16-edge tile per wave; C-fragment lane->col map
// gives 64B-contiguous half-wave gathers/scatters. agg is L2-resident.
// src/dst row offsets hoisted once per tile and reused across all 8 col-tiles.
// ---------------------------------------------------------------------------
__global__ void __launch_bounds__(256) edge_message_kernel(
    const float* __restrict__ ea, const int* __restrict__ srcI,
    const int* __restrict__ dstI, const float* __restrict__ We,
    const float* __restrict__ be, const float* __restrict__ h,
    float* __restrict__ agg, int E) {
  __shared__ float ldsW[F_ * WPE];
  __shared__ float ldsB[D_];
  for (int i = threadIdx.x; i < F_ * D_; i += 256) {
    int r = i >> 7, c = i & 127;
    ldsW[r * WPE + c] = We[i];
  }
  if (threadIdx.x < D_) ldsB[threadIdx.x] = be[threadIdx.x];
  __syncthreads();

  int wave = threadIdx.x >> 5, lane = threadIdx.x & 31;
  int lm = lane & 15, dlt = (lane >> 4) << 1;
  int e0 = (blockIdx.x * 8 + wave) * 16;
  if (e0 >= E) return;
  bool full = (e0 + 16 <= E);              // wave-uniform
  int arow = e0 + lm; if (arow >= E) arow = E - 1;

  v2f A[4];
#pragma unroll
  for (int kt = 0; kt < 4; ++kt) {
    int k = kt * 4 + dlt;
    A[kt].x = ea[arow * F_ + k];
    A[kt].y = ea[arow * F_ + k + 1];
  }
  int mbase = (lane >> 4) << 3;
  // Hoist gather/scatter row offsets (16 edges per tile, 8 rows per lane-half)
  int hoff[8], aoff[8];
#pragma unroll
  for (int v = 0; v < 8; ++v) {
    int e = e0 + mbase + v;
    if (e >= E) e = E - 1;                 // clamped; masked below on partial
    hoff[v] = srcI[e] * D_;
    aoff[v] = dstI[e] * D_;
  }
#pragma unroll
  for (int ct = 0; ct < 8; ++ct) {
    int c = ct * 16 + lm;
    v8f acc = {};
#pragma unroll
    for (int kt = 0; kt < 4; ++kt) {
      int k = kt * 4 + dlt;
      v2f B = { ldsW[k * WPE + c], ldsW[(k + 1) * WPE + c] };
      acc = wmma_f32(A[kt], B, acc);
    }
    float bias = ldsB[c];
    if (full) {
#pragma unroll
      for (int v = 0; v < 8; ++v) {
        float msg = fmaxf(acc[v] + bias + h[hoff[v] + c], 0.0f);
        atomicAdd(&agg[aoff[v] + c], msg);
      }
    } else {
#pragma unroll
      for (int v = 0; v < 8; ++v) {
        if (e0 + mbase + v < E) {
          float msg = fmaxf(acc[v] + bias + h[hoff[v] + c], 0.0f);
          atomicAdd(&agg[aoff[v] + c], msg);
        }
      }
    }
  }
}

// ---------------------------------------------------------------------------
// z1 = ((1+eps)*h + agg) @ W1 + b1   [N,128]@[128,256], + BN1 stat atomics.
// Full W1 (128x256) async-DMA'd to LDS (padded); 128-row activation tile in LDS.
// 8 waves x 16 rows; 16 col-tiles x 32 K-steps = 512 WMMAs per wave.
// Dynamic LDS: 128*WP1 + 128*ZP floats (~198KB of the WGP's 320KB).
// ---------------------------------------------------------------------------
__global__ void __launch_bounds__(256) gin_mlp1_kernel(
    const float* __restrict__ h, const float* __restrict__ agg,
    const float* __restrict__ epsp, const float* __restrict__ W1,
    const float* __restrict__ b1, float* __restrict__ z1,
    float* __restrict__ s1, float* __restrict__ q1, int N) {
  extern __shared__ float lds[];
  float* ldsW = lds;                 // 128 x WP1
  float* ldsZ = lds + 128 * WP1;     // 128 x ZP
  int tid = threadIdx.x;
  int rowBase = blockIdx.x * 128;

  for (int i = tid; i < 128 * 64; i += 256) {        // W1: 128 rows x 64 float4
    int r = i >> 6, c4 = i & 63;
    async_copy16(&ldsW[r * WP1 + c4 * 4], W1 + (r << 8) + (c4 << 2));
  }
  float epsv = 1.0f + epsp[0];
  for (int i = tid; i < 128 * 32; i += 256) {        // zin: 128 rows x 32 float4
    int r = i >> 5, c4 = i & 31;
    int row = rowBase + r;
    float zx = 0.f, zy = 0.f, zz = 0.f, zw = 0.f;
    if (row < N) {
      float4 hv = reinterpret_cast<const float4*>(h)[row * 32 + c4];
      float4 av = reinterpret_cast<const float4*>(agg)[row * 32 + c4];
      zx = epsv * hv.x + av.x; zy = epsv * hv.y + av.y;
      zz = epsv * hv.z + av.z; zw = epsv * hv.w + av.w;
    }
    float* p = &ldsZ[r * ZP + c4 * 4];
    p[0] = zx; p[1] = zy; p[2] = zz; p[3] = zw;
  }
  async_wait0();
  __syncthreads();

  int wave = tid >> 5, lane = tid & 31;
  int lm = lane & 15, dlt = (lane >> 4) << 1;
  int r0 = wave * 16;
  v8f acc[16] = {};
  for (int kt = 0; kt < 32; ++kt) {
    int k = kt * 4 + dlt;
    v2f A = { ldsZ[(r0 + lm) * ZP + k], ldsZ[(r0 + lm) * ZP + k + 1] };
#pragma unroll
    for (int ct = 0; ct < 16; ++ct) {
      v2f B = { ldsW[k * WP1 + ct * 16 + lm], ldsW[(k + 1) * WP1 + ct * 16 + lm] };
      acc[ct] = wmma_f32(A, B, acc[ct]);
    }
  }

  int mbase = (lane >> 4) << 3;
  int rowTop = rowBase + r0;               // wave-uniform
  if (rowTop + 16 <= N) {                  // full stripe: straight-line epilogue
#pragma unroll
    for (int ct = 0; ct < 16; ++ct) {
      int c = ct * 16 + lm;
      float bias = b1[c];
      float ps = 0.f, pq = 0.f;
#pragma unroll
      for (int v = 0; v < 8; ++v) {
        float val = acc[ct][v] + bias;
        z1[(rowTop + mbase + v) * 256 + c] = val;
        ps += val; pq += val * val;
      }
      ps += __shfl_xor(ps, 16, 32);        // combine half-waves (same column)
      pq += __shfl_xor(pq, 16, 32);
      if (lane < 16) { atomicAdd(&s1[c], ps); atomicAdd(&q1[c], pq); }
    }
  } else if (rowTop < N) {                 // partial stripe (rare)
#pragma unroll
    for (int ct = 0; ct < 16; ++ct) {
      int c = ct * 16 + lm;
      float bias = b1[c];
      float ps = 0.f, pq = 0.f;
#pragma unroll
      for (int v = 0; v < 8; ++v) {
        int row = rowTop + mbase + v;
        if (row < N) {
          float val = acc[ct][v] + bias;
          z1[row * 256 + c] = val;
          ps += val; pq += val * val;
        }
      }
      ps += __shfl_xor(ps, 16, 32);
      pq += __shfl_xor(pq, 16, 32);
      if (lane < 16) { atomicAdd(&s1[c], ps); atomicAdd(&q1[c], pq); }
    }
  }
}

// ---------------------------------------------------------------------------
// z2 = relu(bn1(z1)) @ W2 + b2   [N,256]@[256,128], + BN2 stat atomics.
// W2 (256x128) async-DMA'd to LDS; activation staged in two K=128 chunks.
// ---------------------------------------------------------------------------
__global__ void __launch_bounds__(256) gin_mlp2_kernel(
    const float* __restrict__ z1, const float* __restrict__ sc1,
    const float* __restrict__ sh1, const float* __restrict__ W2,
    const float* __restrict__ b2, float* __restrict__ z2,
    float* __restrict__ s2, float* __restrict__ q2, int N) {
  extern __shared__ float lds[];
  float* ldsW = lds;                 // 256 x WP2
  float* ldsZ = lds + 256 * WP2;     // 128 x ZP
  int tid = threadIdx.x;
  int rowBase = blockIdx.x * 128;

  for (int i = tid; i < 256 * 32; i += 256) {        // W2: 256 rows x 32 float4
    int r = i >> 5, c4 = i & 31;
    async_copy16(&ldsW[r * WP2 + c4 * 4], W2 + (r << 7) + (c4 << 2));
  }

  int wave = tid >> 5, lane = tid & 31;
  int lm = lane & 15, dlt = (lane >> 4) << 1;
  int r0 = wave * 16;
  v8f acc[8] = {};

  for (int chunk = 0; chunk < 2; ++chunk) {
    if (chunk == 0) async_wait0();   // W staging landed before first barrier
    __syncthreads();                 // also separates previous chunk's compute
    for (int i = tid; i < 128 * 32; i += 256) {
      int r = i >> 5, c4 = i & 31;
      int row = rowBase + r;
      int cg4 = chunk * 32 + c4;
      float4 zv = make_float4(0.f, 0.f, 0.f, 0.f);
      if (row < N) zv = reinterpret_cast<const float4*>(z1)[row * 64 + cg4];
      float4 sc = reinterpret_cast<const float4*>(sc1)[cg4];
      float4 sh = reinterpret_cast<const float4*>(sh1)[cg4];
      float ax = fmaxf(zv.x * sc.x + sh.x, 0.f);
      float ay = fmaxf(zv.y * sc.y + sh.y, 0.f);
      float az = fmaxf(zv.z * sc.z + sh.z, 0.f);
      float aw = fmaxf(zv.w * sc.w + sh.w, 0.f);
      float* p = &ldsZ[r * ZP + c4 * 4];
      p[0] = ax; p[1] = ay; p[2] = az; p[3] = aw;
    }
    __syncthreads();
    for (int kt = 0; kt < 32; ++kt) {
      int k = kt * 4 + dlt;
      int kw = chunk * 128 + k;
      v2f A = { ldsZ[(r0 + lm) * ZP + k], ldsZ[(r0 + lm) * ZP + k + 1] };
#pragma unroll
      for (int ct = 0; ct < 8; ++ct) {
        v2f B = { ldsW[kw * WP2 + ct * 16 + lm],
                  ldsW[(kw + 1) * WP2 + ct * 16 + lm] };
        acc[ct] = wmma_f32(A, B, acc[ct]);
      }
    }
  }

  int mbase = (lane >> 4) << 3;
  int rowTop = rowBase + r0;
  if (rowTop + 16 <= N) {
#pragma unroll
    for (int ct = 0; ct < 8; ++ct) {
      int c = ct * 16 + lm;
      float bias = b2[c];
      float ps = 0.f, pq = 0.f;
#pragma unroll
      for (int v = 0; v < 8; ++v) {
        float val = acc[ct][v] + bias;
        z2[(rowTop + mbase + v) * D_ + c] = val;
        ps += val; pq += val * val;
      }
      ps += __shfl_xor(ps, 16, 32);
      pq += __shfl_xor(pq, 16, 32);
      if (lane < 16) { atomicAdd(&s2[c], ps); atomicAdd(&q2[c], pq); }
    }
  } else if (rowTop < N) {
#pragma unroll
    for (int ct = 0; ct < 8; ++ct) {
      int c = ct * 16 + lm;
      float bias = b2[c];
      float ps = 0.f, pq = 0.f;
#pragma unroll
      for (int v = 0; v < 8; ++v) {
        int row = rowTop + mbase + v;
        if (row < N) {
          float val = acc[ct][v] + bias;
          z2[row * D_ + c] = val;
          ps += val; pq += val * val;
        }
      }
      ps += __shfl_xor(ps, 16, 32);
      pq += __shfl_xor(pq, 16, 32);
      if (lane < 16) { atomicAdd(&s2[c], ps); atomicAdd(&q2[c], pq); }
    }
  }
}

// BN finalize: scale = g*rsqrt(var+eps), shift = beta - mean*scale
__global__ void bn_finalize_kernel(const float* __restrict__ s,
                                   const float* __restrict__ q,
                                   const float* __restrict__ g,
                                   const float* __restrict__ beta,
                                   float* __restrict__ scale,
                                   float* __restrict__ shift, int C, float invN) {
  int c = blockIdx.x * blockDim.x + threadIdx.x;
  if (c < C) {
    float mean = s[c] * invN;
    float var = q[c] * invN - mean * mean;
    float rstd = rsqrtf(var + 1e-5f);
    float sc = g[c] * rstd;
    scale[c] = sc;
    shift[c] = beta[c] - mean * sc;
  }
}

// h_out = bn2(z2) (optional relu) + h_in   (float4 vectorized)
__global__ void residual_update_kernel(const float* __restrict__ z2,
                                       const float* __restrict__ sc2,
                                       const float* __restrict__ sh2,
                                       const float* hin, float* hout,
                                       int n4, int doRelu) {
  int i = blockIdx.x * blockDim.x + threadIdx.x;
  if (i >= n4) return;
  int c4 = i & 31;
  float4 zv = reinterpret_cast<const float4*>(z2)[i];
  float4 hv = reinterpret_cast<const float4*>(hin)[i];
  float4 sc = reinterpret_cast<const float4*>(sc2)[c4];
  float4 sh = reinterpret_cast<const float4*>(sh2)[c4];
  float ox = zv.x * sc.x + sh.x;
  float oy = zv.y * sc.y + sh.y;
  float oz = zv.z * sc.z + sh.z;
  float ow = zv.w * sc.w + sh.w;
  if (doRelu) {
    ox = fmaxf(ox, 0.f); oy = fmaxf(oy, 0.f);
    oz = fmaxf(oz, 0.f); ow = fmaxf(ow, 0.f);
  }
  float4 o = make_float4(ox + hv.x, oy + hv.y, oz + hv.z, ow + hv.w);
  reinterpret_cast<float4*>(hout)[i] = o;
}

extern "C" void kernel_launch(void* const* d_in, const int* in_sizes, int n_in,
                              void* d_out, int out_size, void* d_ws, size_t ws_size,
                              hipStream_t stream) {
  (void)n_in; (void)out_size; (void)ws_size;
  const float* x   = (const float*)d_in[0];
  const int*   ei  = (const int*)d_in[1];
  const float* ea  = (const float*)d_in[2];
  const float* Wn  = (const float*)d_in[3];
  const float* bn  = (const float*)d_in[4];
  const float* We  = (const float*)d_in[5];
  const float* be  = (const float*)d_in[6];
  const float* eps = (const float*)d_in[7];
  const float* W1  = (const float*)d_in[8];
  const float* b1  = (const float*)d_in[9];
  const float* g1  = (const float*)d_in[10];
  const float* bt1 = (const float*)d_in[11];
  const float* W2  = (const float*)d_in[12];
  const float* b2  = (const float*)d_in[13];
  const float* gb  = (const float*)d_in[14];
  const float* bb  = (const float*)d_in[15];

  int N = in_sizes[0] / F_;
  int E = in_sizes[1] / 2;
  const int* srcI = ei;
  const int* dstI = ei + E;

  // Workspace layout (agg and z2 alias: agg dead before z2 is written)
  float* h    = (float*)d_ws;                       // N*128
  float* z1   = h + (size_t)N * D_;                 // N*256
  float* agg  = z1 + (size_t)N * 2 * D_;            // N*128 (also z2)
  float* z2   = agg;
  float* s1   = agg + (size_t)N * D_;               // 256
  float* q1   = s1 + 256;                           // 256
  float* s2   = q1 + 256;                           // 128
  float* q2   = s2 + 128;                           // 128
  float* sc1  = q2 + 128;                           // 256
  float* sh1  = sc1 + 256;                          // 256
  float* sc2  = sh1 + 256;                          // 128
  float* sh2  = sc2 + 128;                          // 128
  float* out  = (float*)d_out;

  int nodeBlocks = ((N + 15) / 16 + 7) / 8;
  int edgeBlocks = ((E + 15) / 16 + 7) / 8;
  int gemmBlocks = (N + 127) / 128;
  size_t lds1 = (size_t)(128 * WP1 + 128 * ZP) * sizeof(float);
  size_t lds2 = (size_t)(256 * WP2 + 128 * ZP) * sizeof(float);
  hipFuncSetAttribute(reinterpret_cast<const void*>(gin_mlp1_kernel),
                      hipFuncAttributeMaxDynamicSharedMemorySize, (int)lds1);
  hipFuncSetAttribute(reinterpret_cast<const void*>(gin_mlp2_kernel),
                      hipFuncAttributeMaxDynamicSharedMemorySize, (int)lds2);

  float invN = 1.0f / (float)N;
  int n4 = N * (D_ / 4);

  node_enc_kernel<<<nodeBlocks, 256, 0, stream>>>(x, Wn, bn, h, N);

  for (int l = 0; l < L_; ++l) {
    hipMemsetAsync(agg, 0, (size_t)N * D_ * sizeof(float), stream);
    hipMemsetAsync(s1, 0, (256 + 256 + 128 + 128) * sizeof(float), stream);
    edge_message_kernel<<<edgeBlocks, 256, 0, stream>>>(
        ea, srcI, dstI, We + (size_t)l * F_ * D_, be + (size_t)l * D_, h, agg, E);
    gin_mlp1_kernel<<<gemmBlocks, 256, lds1, stream>>>(
        h, agg, eps + l, W1 + (size_t)l * D_ * 2 * D_, b1 + (size_t)l * 2 * D_,
        z1, s1, q1, N);
    bn_finalize_kernel<<<1, 256, 0, stream>>>(s1, q1, g1 + (size_t)l * 2 * D_,
                                              bt1 + (size_t)l * 2 * D_, sc1, sh1,
                                              2 * D_, invN);
    gin_mlp2_kernel<<<gemmBlocks, 256, lds2, stream>>>(
        z1, sc1, sh1, W2 + (size_t)l * 2 * D_ * D_, b2 + (size_t)l * D_,
        z2, s2, q2, N);
    bn_finalize_kernel<<<1, 128, 0, stream>>>(s2, q2, gb + (size_t)l * D_,
                                              bb + (size_t)l * D_, sc2, sh2,
                                              D_, invN);
    float* dest = (l == L_ - 1) ? out : h;
    residual_update_kernel<<<(n4 + 255) / 256, 256, 0, stream>>>(
        z2, sc2, sh2, h, dest, n4, (l < L_ - 1) ? 1 : 0);
  }
}